// Linear_25718264168481
// MI455X (gfx1250) — compile-verified
//
#include <hip/hip_runtime.h>

// OLS forecast == fixed weighted reduction along W, mapped onto
// V_WMMA_F32_16X16X4_F32 (M=16 counties, K=4 time steps, N broadcast).
//
// Shapes fixed by the reference setup: x [256, 64, 3142] f32, horizon=14.

typedef __attribute__((ext_vector_type(2))) float v2f;
typedef __attribute__((ext_vector_type(8))) float v8f;

#define B_DIM   256
#define W_DIM   64
#define C_DIM   3142
#define CCHUNKS ((C_DIM + 15) / 16)      // 197 chunks of 16 counties

// weight[w] = 1/64 + (w - 31.5)*(77 - 31.5)/21840 = -0.05 + w*(45.5/21840)
#define WT_C1  (-0.05f)
#define WT_C2  (45.5f / 21840.0f)

__global__ __launch_bounds__(128)
void ols_forecast_wmma(const float* __restrict__ x, float* __restrict__ out) {
    const int waveInBlock = threadIdx.x >> 5;
    const int g = blockIdx.x * 4 + waveInBlock;            // one wave = one tile
    if (g >= B_DIM * CCHUNKS) return;

    const int b      = g / CCHUNKS;
    const int cChunk = g % CCHUNKS;
    const int c0     = cChunk * 16;

    const int lane = threadIdx.x & 31;
    const int hh   = lane >> 4;        // lane-half: selects K pair {2h, 2h+1}
    const int cm   = lane & 15;        // M index within the 16-county tile

    // Clamp load column for the tail chunk (stores are masked below; EXEC
    // must stay all-1s through the WMMA loop).
    int c = c0 + cm;
    if (c >= C_DIM) c = C_DIM - 1;

    // Column pointer: x[b, 2*hh, c]; row stride is C_DIM floats.
    const float* p = x + (size_t)b * (W_DIM * C_DIM) + (size_t)(2 * hh) * C_DIM + c;

    // B matrix (4x16, weight broadcast over N): lane supplies K = {2h+0, 2h+1}.
    v2f wv;
    wv.x = WT_C1 + WT_C2 * (float)(2 * hh);
    wv.y = wv.x + WT_C2;

    v8f acc = {};

#pragma unroll
    for (int k0 = 0; k0 < W_DIM; k0 += 4) {
        v2f a;
        a.x = p[0];            // x[b, k0 + 2h    , c]  -> A[m=cm, k=2h  ]
        a.y = p[C_DIM];        // x[b, k0 + 2h + 1, c]  -> A[m=cm, k=2h+1]
        // D = A(16x4 f32) * B(4x16 f32) + C ; accumulate over K in 16 steps.
        acc = __builtin_amdgcn_wmma_f32_16x16x4_f32(
            /*neg_a=*/false, a, /*neg_b=*/false, wv,
            /*c_mod=*/(short)0, acc, /*reuse_a=*/false, /*reuse_b=*/false);
        p += (size_t)4 * C_DIM;
        wv.x += 4.0f * WT_C2;
        wv.y += 4.0f * WT_C2;
    }

    // D layout: lane 0 holds out[c0+0..7] in acc[0..7]; lane 16 holds
    // out[c0+8..15]. All N columns are identical, so two lanes store the tile.
    if (cm == 0) {
        const int cb = c0 + hh * 8;
        float* o = out + (size_t)b * C_DIM + cb;
        union { v8f v; float f[8]; } u;
        u.v = acc;
        if (cb + 8 <= C_DIM) {
            // 8B-aligned (row byte-stride 12568 and cb are multiples of 8B):
            ((float2*)o)[0] = make_float2(u.f[0], u.f[1]);
            ((float2*)o)[1] = make_float2(u.f[2], u.f[3]);
            ((float2*)o)[2] = make_float2(u.f[4], u.f[5]);
            ((float2*)o)[3] = make_float2(u.f[6], u.f[7]);
        } else {
#pragma unroll
            for (int i = 0; i < 8; ++i)
                if (cb + i < C_DIM) o[i] = u.f[i];
        }
    }
}

extern "C" void kernel_launch(void* const* d_in, const int* in_sizes, int n_in,
                              void* d_out, int out_size, void* d_ws, size_t ws_size,
                              hipStream_t stream) {
    (void)in_sizes; (void)n_in; (void)d_ws; (void)ws_size; (void)out_size;
    const float* x = (const float*)d_in[0];   // [256, 64, 3142] f32
    float* out = (float*)d_out;               // [256, 3142] f32

    const int totalWaves = B_DIM * CCHUNKS;   // 50432
    const int blocks = (totalWaves + 3) / 4;  // 4 waves (128 threads) per block
    ols_forecast_wmma<<<blocks, 128, 0, stream>>>(x, out);
}